// Model_25056839205321
// MI455X (gfx1250) — compile-verified
//
#include <hip/hip_runtime.h>
#include <stdint.h>

#define NTOK  8192
#define HID   2560
#define HC    4
#define CPT   320                 // chunks per token row = HID / 8 bf16
#define GRID  1024                // fixed launch grid
#define NITER (NTOK / GRID)       // 8 tokens per block, exact
#define NBUF  3                   // triple buffer -> prefetch depth 2
#define ROWB  (CPT * 16)          // 5120 B per row (x or one residual j)
#define BUFB  (5 * ROWB)          // 25600 B per token buffer

typedef uint32_t u32x4 __attribute__((ext_vector_type(4)));
typedef float    f32x2 __attribute__((ext_vector_type(2)));

// ---- CDNA5 async copy: global -> LDS, ASYNCcnt-tracked, NT (stream-once) ----
__device__ __forceinline__ void async_b128(uint32_t lds_off, const void* gptr) {
    asm volatile("global_load_async_to_lds_b128 %0, %1, off th:TH_LOAD_NT"
                 :: "v"(lds_off), "v"(gptr) : "memory");
}
// INST_OFFSET adds to BOTH the LDS and global address (ISA 08 §4.4), so rows with
// matching pitch on both sides share one address pair.
template <int OFF>
__device__ __forceinline__ void async_b128_off(uint32_t lds_off, const void* gptr) {
    asm volatile("global_load_async_to_lds_b128 %0, %1, off offset:%2 th:TH_LOAD_NT"
                 :: "v"(lds_off), "v"(gptr), "n"(OFF) : "memory");
}

// 5 async ops per wave: x row -> row0, residual j=0..3 -> rows 1..4
__device__ __forceinline__ void fill_token(uint32_t lds_buf, const char* gx, const char* gr) {
    async_b128(lds_buf, gx);                       // x
    async_b128(lds_buf + ROWB, gr);                // j=0
    async_b128_off<1 * ROWB>(lds_buf + ROWB, gr);  // j=1 (offset shifts LDS too)
    async_b128_off<2 * ROWB>(lds_buf + ROWB, gr);  // j=2
    async_b128_off<3 * ROWB>(lds_buf + ROWB, gr);  // j=3
}

// two bf16 (packed in a dword) -> f32x2
__device__ __forceinline__ f32x2 bf2f(uint32_t u) {
    f32x2 r;
    r.x = __uint_as_float(u << 16);
    r.y = __uint_as_float(u & 0xffff0000u);
    return r;
}
__device__ __forceinline__ f32x2 splat2(float v) { f32x2 r; r.x = v; r.y = v; return r; }

// f32x2 -> packed bf16 pair in one VALU op (RNE, matches astype(bf16))
__device__ __forceinline__ uint32_t fpackbf(f32x2 v) {
    uint32_t r;
    asm("v_cvt_pk_bf16_f32 %0, %1, %2" : "=v"(r) : "v"(v.x), "v"(v.y));
    return r;
}

__global__ __launch_bounds__(CPT) void hc_mix_kernel(
    const u32x4* __restrict__ xg,    // bf16 [NTOK][HID]
    const u32x4* __restrict__ rg,    // bf16 [NTOK][HC][HID]
    const float* __restrict__ plm,   // f32  [NTOK][HC]
    const float* __restrict__ comb,  // f32  [NTOK][HC][HC]  comb[t][j][i]
    u32x4* __restrict__ outg)        // bf16 [NTOK][HC][HID]
{
    __shared__ u32x4 smem[NBUF][5][CPT];   // 75 KB of the 320 KB/WGP

    const int c  = threadIdx.x;
    const int t0 = blockIdx.x;
    const uint32_t c16  = (uint32_t)c * 16u;
    const uint32_t lds0 = (uint32_t)(uintptr_t)(&smem[0][0][0]) + c16;

    // fill-side pointers, advanced by constant strides (strength-reduced)
    const char* gx = (const char*)xg + (size_t)t0 * ROWB + c16;
    const char* gr = (const char*)rg + (size_t)t0 * (HC * ROWB) + c16;
    const size_t gx_step = (size_t)GRID * ROWB;
    const size_t gr_step = (size_t)GRID * (HC * ROWB);

    // depth-2 prefetch prologue (NITER >= 2 by construction)
    fill_token(lds0, gx, gr);            gx += gx_step; gr += gr_step;
    fill_token(lds0 + BUFB, gx, gr);     gx += gx_step; gr += gr_step;

    char* op = (char*)outg + (size_t)t0 * (HC * ROWB) + c16;

    uint32_t fill_off = 2 * BUFB;   // LDS offset of buffer to fill next
    int bc = 0;                     // buffer to consume
    for (int it = 0; it < NITER; ++it) {
        const int t = t0 + it * GRID;

        if (it < NITER - 2) {
            fill_token(lds0 + fill_off, gx, gr);
            gx += gx_step; gr += gr_step;
            fill_off = (fill_off == 2 * BUFB) ? 0u : fill_off + BUFB;
        }

        // per-token mix weights: VMEM latency overlaps the async wait
        float p[HC], w[HC][HC];
        #pragma unroll
        for (int i = 0; i < HC; ++i) p[i] = plm[(size_t)t * HC + i];
        #pragma unroll
        for (int j = 0; j < HC; ++j)
            #pragma unroll
            for (int i = 0; i < HC; ++i)
                w[j][i] = comb[(size_t)t * (HC * HC) + j * HC + i];

        // async loads retire in order; 5 ops per outstanding token per wave
        if (it < NITER - 2)       asm volatile("s_wait_asynccnt 10" ::: "memory");
        else if (it == NITER - 2) asm volatile("s_wait_asynccnt 5"  ::: "memory");
        else                      asm volatile("s_wait_asynccnt 0"  ::: "memory");
        __syncthreads();  // all waves' fills of smem[bc] visible

        const u32x4 xq = smem[bc][0][c];
        f32x2 xf[4];
        xf[0] = bf2f(xq.x); xf[1] = bf2f(xq.y);
        xf[2] = bf2f(xq.z); xf[3] = bf2f(xq.w);

        f32x2 acc[HC][4];
        #pragma unroll
        for (int i = 0; i < HC; ++i) {
            const f32x2 pi = splat2(p[i]);
            #pragma unroll
            for (int q = 0; q < 4; ++q)
                acc[i][q] = pi * xf[q];
        }

        #pragma unroll
        for (int j = 0; j < HC; ++j) {
            const u32x4 rq = smem[bc][1 + j][c];
            f32x2 rf[4];
            rf[0] = bf2f(rq.x); rf[1] = bf2f(rq.y);
            rf[2] = bf2f(rq.z); rf[3] = bf2f(rq.w);
            #pragma unroll
            for (int i = 0; i < HC; ++i) {
                const f32x2 wj = splat2(w[j][i]);
                #pragma unroll
                for (int q = 0; q < 4; ++q)
                    acc[i][q] = __builtin_elementwise_fma(wj, rf[q], acc[i][q]);
            }
        }

        #pragma unroll
        for (int i = 0; i < HC; ++i) {
            u32x4 o;
            o.x = fpackbf(acc[i][0]);
            o.y = fpackbf(acc[i][1]);
            o.z = fpackbf(acc[i][2]);
            o.w = fpackbf(acc[i][3]);
            __builtin_nontemporal_store(o, (u32x4*)(op + (size_t)i * ROWB));
        }
        op += gr_step;

        __syncthreads();  // finish reading smem[bc] before it is refilled
        bc = (bc == NBUF - 1) ? 0 : bc + 1;
    }
}

extern "C" void kernel_launch(void* const* d_in, const int* in_sizes, int n_in,
                              void* d_out, int out_size, void* d_ws, size_t ws_size,
                              hipStream_t stream) {
    (void)in_sizes; (void)n_in; (void)out_size; (void)d_ws; (void)ws_size;
    const u32x4* xg   = (const u32x4*)d_in[0];   // bf16 [8192][2560]
    const u32x4* rg   = (const u32x4*)d_in[1];   // bf16 [8192][4][2560]
    const float* plm  = (const float*)d_in[2];   // f32  [8192][4][1]
    const float* comb = (const float*)d_in[3];   // f32  [8192][4][4]
    u32x4* outg       = (u32x4*)d_out;           // bf16 [8192][4][2560]

    hc_mix_kernel<<<dim3(GRID), dim3(CPT), 0, stream>>>(xg, rg, plm, comb, outg);
}